// ExpertConv2d_62371515073071
// MI455X (gfx1250) — compile-verified
//
#include <hip/hip_runtime.h>
#include <hip/hip_bf16.h>

typedef __attribute__((ext_vector_type(2))) float v2f;
typedef __attribute__((ext_vector_type(8))) float v8f;

#define NPATCH   256          // B * 8 * 8
#define AGGW_PER 36864        // 9 * 64 * 64 floats per patch
#define PAIR_STRIDE 336       // padded 18*18=324 -> 336 so 2*336 % 64 == 32 (no bank conflicts)
#define XS_FLOATS (32 * PAIR_STRIDE * 2)   // 21504 floats = 86,016 B

// ---------------------------------------------------------------------------
// Kernel 1: premix expert weights/bias per patch.
//   aggW[p][t][co][ci]  (ci innermost -> contiguous float2 A-fragment loads)
//   aggB[p][co]
// ---------------------------------------------------------------------------
__global__ __launch_bounds__(256) void expertconv_mix_kernel(
    const float* __restrict__ v, const float* __restrict__ weight,
    const float* __restrict__ bias, float* __restrict__ aggW,
    float* __restrict__ aggB) {
  const int p = blockIdx.x;
  const int bb = p >> 6;
  const int gy = (p >> 3) & 7;
  const int gx = p & 7;

  float vv[5];
#pragma unroll
  for (int k = 0; k < 5; ++k)
    vv[k] = v[((bb * 5 + k) * 8 + gy) * 8 + gx];

  const int tid = threadIdx.x;
  if (tid < 64) {
    float s = 0.0f;
#pragma unroll
    for (int k = 0; k < 5; ++k) s += vv[k] * bias[k * 64 + tid];
    aggB[p * 64 + tid] = s;
  }

  for (int idx = tid; idx < AGGW_PER; idx += 256) {
    const int t = idx >> 12;         // tap 0..8
    const int rem = idx & 4095;      // co*64 + ci
    float s = 0.0f;
#pragma unroll
    for (int k = 0; k < 5; ++k)
      s += vv[k] * weight[(k * 4096 + rem) * 9 + t];
    aggW[(size_t)p * AGGW_PER + idx] = s;
  }
}

// ---------------------------------------------------------------------------
// Kernel 2: per-patch implicit-GEMM conv via V_WMMA_F32_16X16X4_F32.
// grid = (16 spatial blocks, 256 patches), block = 256 threads = 8 waves.
// Each block: out[co=0..63, 16x16 pixel tile] of one patch.
// LDS x tile layout: ci-pair interleaved -> B-fragment = one ds_load_b64:
//   xs[(cipair*336 + iy*18 + ix)*2 + (ci&1)], cipair plane stride padded to
//   336 float pairs (672 floats) so lane halves (K pair 0/1 vs 2/3) hit disjoint bank halves.
// ---------------------------------------------------------------------------
__global__ __launch_bounds__(256) void expertconv_wmma_kernel(
    const float* __restrict__ x, const float* __restrict__ aggW,
    const float* __restrict__ aggB, float* __restrict__ out) {
  extern __shared__ float xs[];   // 86,016 B
  __shared__ float sbias[64];

  const int p = blockIdx.y;
  const int bb = p >> 6;
  const int gy = (p >> 3) & 7;
  const int gx = p & 7;
  const int by = blockIdx.x >> 2;
  const int bx = blockIdx.x & 3;
  const int oy0 = by * 16;
  const int ox0 = bx * 16;

  const int tid = threadIdx.x;
  const size_t YX = (size_t)512 * 512;

  if (tid < 64) sbias[tid] = aggB[p * 64 + tid];

  // Stage input tile (with per-patch zero halo) into LDS, pair-interleaved.
  const float* xpatch =
      x + (size_t)bb * 64 * YX + (size_t)(gy * 64) * 512 + (gx * 64);
  for (int idx = tid; idx < 64 * 18 * 18; idx += 256) {
    const int ci = idx / 324;
    const int r = idx - ci * 324;
    const int iy = r / 18;
    const int ix = r - iy * 18;
    const int ly = oy0 - 1 + iy;   // local-in-patch coords (pad=1 per patch)
    const int lx = ox0 - 1 + ix;
    float val = 0.0f;
    if (ly >= 0 && ly < 64 && lx >= 0 && lx < 64)
      val = xpatch[(size_t)ci * YX + (size_t)ly * 512 + lx];
    xs[(((ci >> 1) * PAIR_STRIDE + iy * 18 + ix) << 1) + (ci & 1)] = val;
  }
  __syncthreads();

  const int lane = tid & 31;
  const int w = tid >> 5;
  const int wm = w >> 2;        // 0..1 : co rows [wm*32, wm*32+32)
  const int wn = w & 3;         // 0..3 : pixel rows [wn*4, wn*4+4)
  const int col = lane & 15;    // A: M row / B: N col
  const int khalf = lane >> 4;  // 0/1 -> K pair {0,1} or {2,3}

  v8f acc[2][4];
  {
    v8f z = {};
#pragma unroll
    for (int mi = 0; mi < 2; ++mi)
#pragma unroll
      for (int ni = 0; ni < 4; ++ni) acc[mi][ni] = z;
  }

  const float* wp = aggW + (size_t)p * AGGW_PER;

  for (int t = 0; t < 9; ++t) {
    const int du = t / 3;
    const int dv = t - du * 3;
    const float* wt = wp + t * 4096;
    // Pull the next tap's 16KB weight panel toward the WGP while computing.
    __builtin_prefetch(wt + 4096 + (tid << 4), 0, 1);
    for (int kb = 0; kb < 16; ++kb) {
      const int cipair = kb * 2 + khalf;     // selects K pair {0,1}/{2,3}
      const int kbase = cipair * 2;
      v2f a[2];
#pragma unroll
      for (int mi = 0; mi < 2; ++mi) {
        const int co = wm * 32 + mi * 16 + col;
        const v2f* ap = (const v2f*)(wt + co * 64 + kbase);  // global b64
        a[mi] = *ap;
      }
#pragma unroll
      for (int ni = 0; ni < 4; ++ni) {
        const int py = wn * 4 + ni;
        const int iy = py + du;          // halo index 0..17
        const int ix = col + dv;         // halo index 0..17
        const v2f bv =
            *(const v2f*)&xs[(cipair * PAIR_STRIDE + iy * 18 + ix) << 1];
#pragma unroll
        for (int mi = 0; mi < 2; ++mi) {
          acc[mi][ni] = __builtin_amdgcn_wmma_f32_16x16x4_f32(
              false, a[mi], false, bv, (short)0, acc[mi][ni], false, false);
        }
      }
    }
  }

  // Write back: C/D layout -> co = base + khalf*8 + r, pixel col = lane&15.
  float* outBase = out + (size_t)bb * 64 * YX +
                   (size_t)(gy * 64 + oy0) * 512 + (gx * 64 + ox0);
#pragma unroll
  for (int mi = 0; mi < 2; ++mi) {
#pragma unroll
    for (int ni = 0; ni < 4; ++ni) {
      const int py = wn * 4 + ni;
#pragma unroll
      for (int r = 0; r < 8; ++r) {
        const int co = wm * 32 + mi * 16 + khalf * 8 + r;
        outBase[(size_t)co * YX + (size_t)py * 512 + col] =
            acc[mi][ni][r] + sbias[co];
      }
    }
  }
}

extern "C" void kernel_launch(void* const* d_in, const int* in_sizes, int n_in,
                              void* d_out, int out_size, void* d_ws,
                              size_t ws_size, hipStream_t stream) {
  (void)in_sizes; (void)n_in; (void)out_size; (void)ws_size;
  const float* x = (const float*)d_in[0];
  const float* v = (const float*)d_in[1];
  const float* weight = (const float*)d_in[2];
  const float* bias = (const float*)d_in[3];
  float* out = (float*)d_out;

  float* aggW = (float*)d_ws;                       // 256*36864 f32 = 37.7 MB
  float* aggB = aggW + (size_t)NPATCH * AGGW_PER;   // 256*64 f32

  expertconv_mix_kernel<<<NPATCH, 256, 0, stream>>>(v, weight, bias, aggW,
                                                    aggB);

  dim3 grid(16, NPATCH);
  const size_t shmem = (size_t)XS_FLOATS * sizeof(float);  // 86,016 B dynamic
  expertconv_wmma_kernel<<<grid, 256, shmem, stream>>>(x, aggW, aggB, out);
}